// GNNLayer_75763223102025
// MI455X (gfx1250) — compile-verified
//
#include <hip/hip_runtime.h>

typedef __attribute__((ext_vector_type(2))) float v2f;
typedef __attribute__((ext_vector_type(4))) float v4f;
typedef __attribute__((ext_vector_type(8))) float v8f;

#define IN_F 512
#define OUT_F 128
#define NEG_SLOPE 0.2f

// ---------------------------------------------------------------------------
// support = features @ weight  via V_WMMA_F32_16X16X4_F32 (exact fp32)
// Block = 256 threads = 8 waves; block owns 64 rows x 128 cols.
// Wave w owns cols [16w,16w+16) and FOUR 16-row M-tiles -> one B-tile load
// feeds 4 WMMAs (1.5 vmem ops per WMMA instead of 3).
// A 16x4 f32 layout (ISA 7.12.2): VGPR0 = {lanes0-15: K=k, lanes16-31: K=k+2},
//                                 VGPR1 = {K=k+1, K=k+3}  -> one float2 load/lane.
// C/D 16x16 f32: VGPR j -> row mt + j + 8*(lane>=16), col n0 + (lane&15).
// ---------------------------------------------------------------------------
__global__ __launch_bounds__(256) void gemm_wmma_f32(const float* __restrict__ A,
                                                     const float* __restrict__ W,
                                                     float* __restrict__ S,
                                                     int nrows) {
    const int wave  = threadIdx.x >> 5;
    const int lane  = threadIdx.x & 31;
    const int lhalf = lane >> 4;      // 0: lanes 0-15, 1: lanes 16-31
    const int l16   = lane & 15;
    const int m0    = blockIdx.x << 6;   // 64 rows per block
    const int n0    = wave << 4;

    const float* aptr[4];
#pragma unroll
    for (int t = 0; t < 4; ++t) {
        int r = m0 + t * 16 + l16;
        if (r >= nrows) r = nrows - 1;             // clamp loads for tail block
        aptr[t] = A + (size_t)r * IN_F + (lhalf << 1);
    }
    const float* wptr = W + (size_t)(lhalf << 1) * OUT_F + n0 + l16;

    v8f c0 = {}, c1 = {}, c2 = {}, c3 = {};
#pragma unroll 2
    for (int k = 0; k < IN_F; k += 4) {
        v2f b;
        b.x = wptr[k * OUT_F];                     // K = k + 2h
        b.y = wptr[k * OUT_F + OUT_F];             // K = k + 1 + 2h
        v2f a0 = *(const v2f*)(aptr[0] + k);
        v2f a1 = *(const v2f*)(aptr[1] + k);
        v2f a2 = *(const v2f*)(aptr[2] + k);
        v2f a3 = *(const v2f*)(aptr[3] + k);
        c0 = __builtin_amdgcn_wmma_f32_16x16x4_f32(false, a0, false, b, (short)0, c0, false, false);
        c1 = __builtin_amdgcn_wmma_f32_16x16x4_f32(false, a1, false, b, (short)0, c1, false, false);
        c2 = __builtin_amdgcn_wmma_f32_16x16x4_f32(false, a2, false, b, (short)0, c2, false, false);
        c3 = __builtin_amdgcn_wmma_f32_16x16x4_f32(false, a3, false, b, (short)0, c3, false, false);
    }

    v8f acc[4] = {c0, c1, c2, c3};
    if (m0 + 64 <= nrows) {
        // fast path: whole 64-row block in range -> unconditional stores
#pragma unroll
        for (int t = 0; t < 4; ++t) {
            float* sp = S + (size_t)(m0 + t * 16 + (lhalf << 3)) * OUT_F + n0 + l16;
#pragma unroll
            for (int j = 0; j < 8; ++j) sp[j * OUT_F] = acc[t][j];
        }
    } else {
        // tail block: per-row guard (rare path)
#pragma unroll
        for (int t = 0; t < 4; ++t) {
            const int rbase = m0 + t * 16 + (lhalf << 3);
            float* sp = S + (size_t)rbase * OUT_F + n0 + l16;
#pragma unroll
            for (int j = 0; j < 8; ++j) {
                if (rbase + j < nrows) sp[j * OUT_F] = acc[t][j];
            }
        }
    }
}

// ---------------------------------------------------------------------------
// One wave per edge: gather 128 floats of support[col] (L2-resident),
// scale by edge value, atomically accumulate into out[row].
// ---------------------------------------------------------------------------
__global__ __launch_bounds__(256) void edge_scatter(const int*   __restrict__ rows,
                                                    const int*   __restrict__ cols,
                                                    const float* __restrict__ vals,
                                                    const float* __restrict__ S,
                                                    float*       __restrict__ out,
                                                    int nedges) {
    const int lane = threadIdx.x & 31;
    const int e = (int)((blockIdx.x * blockDim.x + threadIdx.x) >> 5);
    if (e >= nedges) return;

    const int   r = rows[e];
    const int   c = cols[e];
    const float v = vals[e];

    const v4f s = *(const v4f*)(S + (size_t)c * OUT_F + (lane << 2));
    float* op = out + (size_t)r * OUT_F + (lane << 2);
    atomicAdd(op + 0, v * s.x);
    atomicAdd(op + 1, v * s.y);
    atomicAdd(op + 2, v * s.z);
    atomicAdd(op + 3, v * s.w);
}

// ---------------------------------------------------------------------------
__global__ void zero_f32(float* __restrict__ p, int n) {
    int i = (int)(blockIdx.x * blockDim.x + threadIdx.x) * 4;
    if (i + 3 < n) {
        *(v4f*)(p + i) = (v4f){0.f, 0.f, 0.f, 0.f};
    } else {
        for (; i < n; ++i) p[i] = 0.f;
    }
}

__global__ void leaky_relu_inplace(float* __restrict__ p, int n) {
    int i = (int)(blockIdx.x * blockDim.x + threadIdx.x) * 4;
    if (i + 3 < n) {
        v4f x = *(const v4f*)(p + i);
        x.x = (x.x >= 0.f) ? x.x : NEG_SLOPE * x.x;
        x.y = (x.y >= 0.f) ? x.y : NEG_SLOPE * x.y;
        x.z = (x.z >= 0.f) ? x.z : NEG_SLOPE * x.z;
        x.w = (x.w >= 0.f) ? x.w : NEG_SLOPE * x.w;
        *(v4f*)(p + i) = x;
    } else {
        for (; i < n; ++i) {
            float x = p[i];
            p[i] = (x >= 0.f) ? x : NEG_SLOPE * x;
        }
    }
}

// ---------------------------------------------------------------------------
extern "C" void kernel_launch(void* const* d_in, const int* in_sizes, int n_in,
                              void* d_out, int out_size, void* d_ws, size_t ws_size,
                              hipStream_t stream) {
    const float* features = (const float*)d_in[0];   // [N, 512] f32
    const float* weight   = (const float*)d_in[1];   // [512, 128] f32
    const int*   rows     = (const int*)  d_in[2];   // [E] i32
    const int*   cols     = (const int*)  d_in[3];   // [E] i32
    const float* vals     = (const float*)d_in[4];   // [E] f32

    float* out     = (float*)d_out;                  // [N, 128] f32
    float* support = (float*)d_ws;                   // [N, 128] f32 scratch (51.2 MB)

    const int nrows  = in_sizes[0] / IN_F;           // 100000
    const int nedges = in_sizes[2];                  // 1600000
    const int nout   = out_size;                     // nrows * 128

    // 1) zero accumulator (harness poisons d_out)
    zero_f32<<<(nout + 1023) / 1024, 256, 0, stream>>>(out, nout);

    // 2) dense projection on the WMMA pipe -> scratch (stays L2-resident)
    gemm_wmma_f32<<<(nrows + 63) / 64, 256, 0, stream>>>(features, weight, support, nrows);

    // 3) edge gather/scale/atomic-scatter (1 wave per edge, 8 edges per block)
    edge_scatter<<<(nedges + 7) / 8, 256, 0, stream>>>(rows, cols, vals, support, out, nedges);

    // 4) leaky_relu(0.2) in place
    leaky_relu_inplace<<<(nout + 1023) / 1024, 256, 0, stream>>>(out, nout);
}